// LocalWLGNN_30116310679891
// MI455X (gfx1250) — compile-verified
//
#include <hip/hip_runtime.h>
#include <hip/hip_bf16.h>

typedef __attribute__((ext_vector_type(2))) float v2f;
typedef __attribute__((ext_vector_type(8))) float v8f;

// ---------------------------------------------------------------------------
// Kernel 1: zero the scatter-destination column blocks feats[:, dim_inner..dim_cat)
// (must run every call; workspace is not re-poisoned between graph replays)
// ---------------------------------------------------------------------------
__global__ __launch_bounds__(256) void zero_tail_kernel(float* __restrict__ feats,
                                                        int n_nodes, int dim_inner,
                                                        int dim_cat) {
    const int per_row4 = (dim_cat - dim_inner) >> 2;          // float4s per row tail
    size_t idx = (size_t)blockIdx.x * blockDim.x + threadIdx.x;
    size_t row = idx / per_row4;
    int c4 = (int)(idx % per_row4);
    if (row >= (size_t)n_nodes) return;
    float4* p = (float4*)(feats + row * (size_t)dim_cat + dim_inner);
    p[c4] = make_float4(0.f, 0.f, 0.f, 0.f);
}

// ---------------------------------------------------------------------------
// Kernel 2: x_lin = X @ W0 + b0  ->  feats[:, 0:dim_inner)   (row stride dim_cat)
// One wave32 computes a 16x16 tile with V_WMMA_F32_16X16X4_F32, K in steps of 4.
// Block = 256 threads = 8 waves; wave w owns N-tile w (covers dim_inner = 128).
// A layout (16x4 f32): lane<16 -> M=lane, K=0 (v0)/K=1 (v1); lane>=16 -> K=2/K=3.
// B layout (4x16):     lane<16 -> K=0 (v0)/K=1 (v1), N=lane; lane>=16 -> K=2/K=3.
// C/D layout: VGPR r -> M=r (lanes 0-15), M=r+8 (lanes 16-31), N=lane&15.
// ---------------------------------------------------------------------------
__global__ __launch_bounds__(256) void gemm1_wmma_kernel(const float* __restrict__ X,
                                                         const float* __restrict__ W0,
                                                         const float* __restrict__ b0,
                                                         float* __restrict__ feats,
                                                         int n_nodes, int dim_in,
                                                         int dim_inner, int dim_cat) {
    const int wave = threadIdx.x >> 5;
    const int lane = threadIdx.x & 31;
    const int half = lane >> 4;           // 0: lanes 0-15, 1: lanes 16-31
    const int l    = lane & 15;
    const int m0   = blockIdx.x * 16;
    const int n0   = wave * 16;
    if (m0 >= n_nodes || n0 >= dim_inner) return;

    int arow = m0 + l;
    if (arow > n_nodes - 1) arow = n_nodes - 1;               // clamp (junk rows unwritten)
    const float* __restrict__ A = X + (size_t)arow * dim_in;

    v8f acc = {};
    for (int k = 0; k < dim_in; k += 4) {
        const int ka = k + 2 * half;
        v2f a, b;
        a.x = A[ka + 0];
        a.y = A[ka + 1];
        b.x = W0[(size_t)(ka + 0) * dim_inner + n0 + l];
        b.y = W0[(size_t)(ka + 1) * dim_inner + n0 + l];
        acc = __builtin_amdgcn_wmma_f32_16x16x4_f32(
            /*neg_a=*/false, a, /*neg_b=*/false, b,
            /*c_mod=*/(short)0, acc, /*reuse_a=*/false, /*reuse_b=*/false);
    }

    const float bias = b0[n0 + l];
    #pragma unroll
    for (int r = 0; r < 8; ++r) {
        const int row = m0 + r + 8 * half;
        if (row < n_nodes)
            feats[(size_t)row * dim_cat + (n0 + l)] = acc[r] + bias;
    }
}

// ---------------------------------------------------------------------------
// Kernel 3: one hop of gather + scatter-add.
// 32 lanes per edge; each lane moves 4 consecutive floats (128 = 32*4).
// Source/dest column-blocks stay L2-resident (153.6 MB buffer < 192 MB L2),
// so the float atomics resolve in L2.
// ---------------------------------------------------------------------------
__global__ __launch_bounds__(256) void edge_scatter_kernel(const float* __restrict__ src_blk,
                                                           float* __restrict__ dst_blk,
                                                           const int* __restrict__ src_idx,
                                                           const int* __restrict__ dst_idx,
                                                           int n_edges, int dim_inner,
                                                           int dim_cat) {
    long long t = (long long)blockIdx.x * blockDim.x + threadIdx.x;
    int edge = (int)(t >> 5);
    int lane = (int)(t & 31);
    if (edge >= n_edges) return;
    const int c = lane * 4;
    if (c >= dim_inner) return;

    const int s = src_idx[edge];
    const int d = dst_idx[edge];
    float4 v = *(const float4*)(src_blk + (size_t)s * dim_cat + c);
    float* drow = dst_blk + (size_t)d * dim_cat + c;
    atomicAdd(drow + 0, v.x);
    atomicAdd(drow + 1, v.y);
    atomicAdd(drow + 2, v.z);
    atomicAdd(drow + 3, v.w);
}

// ---------------------------------------------------------------------------
// Kernel 4: head, fused with the label gather.
// pred[i] = concat_row(node_label_index[i]) @ Wp + bp, with the first
// dim_inner columns scaled by (1+eps) (the reference's "out" block).
// Also emits node_label (as float) into the tail of d_out.
// ---------------------------------------------------------------------------
__global__ __launch_bounds__(256) void head_kernel(const float* __restrict__ feats,
                                                   const float* __restrict__ Wp,
                                                   const float* __restrict__ bp,
                                                   const int* __restrict__ lab_idx,
                                                   const int* __restrict__ labels,
                                                   const float* __restrict__ eps,
                                                   float* __restrict__ out,
                                                   int n_labels, int dim_inner,
                                                   int dim_cat, int dim_out) {
    long long t = (long long)blockIdx.x * blockDim.x + threadIdx.x;
    int i = (int)(t / dim_out);
    int j = (int)(t % dim_out);
    if (i >= n_labels) return;

    const float scale = 1.0f + eps[0];
    const int node = lab_idx[i];
    const float* __restrict__ row = feats + (size_t)node * dim_cat;

    float acc = bp[j];
    for (int k = 0; k < dim_inner; ++k)
        acc = fmaf(scale * row[k], Wp[(size_t)k * dim_out + j], acc);
    for (int k = dim_inner; k < dim_cat; ++k)
        acc = fmaf(row[k], Wp[(size_t)k * dim_out + j], acc);

    out[(size_t)i * dim_out + j] = acc;
    if (j == 0)
        out[(size_t)n_labels * dim_out + i] = (float)labels[i];
}

// ---------------------------------------------------------------------------
extern "C" void kernel_launch(void* const* d_in, const int* in_sizes, int n_in,
                              void* d_out, int out_size, void* d_ws, size_t ws_size,
                              hipStream_t stream) {
    const float* x          = (const float*)d_in[0];
    const float* W0         = (const float*)d_in[1];
    const float* b0         = (const float*)d_in[2];
    const float* eps        = (const float*)d_in[3];
    const float* Wp         = (const float*)d_in[4];
    const float* bp         = (const float*)d_in[5];
    const int*   agg_sc     = (const int*)d_in[6];
    const int*   agg_ni     = (const int*)d_in[7];
    const int*   lab_idx    = (const int*)d_in[8];
    const int*   labels     = (const int*)d_in[9];
    float*       out        = (float*)d_out;

    const int dim_inner = in_sizes[2];                 // 128
    const int dim_in    = in_sizes[1] / dim_inner;     // 128
    const int n_nodes   = in_sizes[0] / dim_in;        // 100000
    const int dim_out   = in_sizes[5];                 // 40
    const int dim_cat   = in_sizes[4] / dim_out;       // 384
    const int hops      = dim_cat / dim_inner - 1;     // 2
    const int n_edges   = in_sizes[6] / hops;          // 1600000
    const int n_labels  = in_sizes[8];                 // 10000

    float* feats = (float*)d_ws;                       // [n_nodes, dim_cat] f32

    // 1. zero the hop-accumulation column blocks
    {
        size_t total4 = (size_t)n_nodes * (size_t)((dim_cat - dim_inner) >> 2);
        unsigned blocks = (unsigned)((total4 + 255) / 256);
        zero_tail_kernel<<<dim3(blocks), dim3(256), 0, stream>>>(
            feats, n_nodes, dim_inner, dim_cat);
    }

    // 2. x_lin = X @ W0 + b0 into feats[:, 0:dim_inner)  (WMMA f32)
    {
        unsigned blocks = (unsigned)((n_nodes + 15) / 16);
        gemm1_wmma_kernel<<<dim3(blocks), dim3(256), 0, stream>>>(
            x, W0, b0, feats, n_nodes, dim_in, dim_inner, dim_cat);
    }

    // 3. hops: gather from block h, atomic-accumulate into block h+1
    for (int hop = 0; hop < hops; ++hop) {
        const int* s = agg_sc + (size_t)hop * n_edges;
        const int* d = agg_ni + (size_t)hop * n_edges;
        const float* src_blk = feats + (size_t)hop * dim_inner;
        float*       dst_blk = feats + (size_t)(hop + 1) * dim_inner;
        long long total = (long long)n_edges * 32;
        unsigned blocks = (unsigned)((total + 255) / 256);
        edge_scatter_kernel<<<dim3(blocks), dim3(256), 0, stream>>>(
            src_blk, dst_blk, s, d, n_edges, dim_inner, dim_cat);
    }

    // 4. head on labeled rows only + label passthrough
    {
        long long total = (long long)n_labels * dim_out;
        unsigned blocks = (unsigned)((total + 255) / 256);
        head_kernel<<<dim3(blocks), dim3(256), 0, stream>>>(
            feats, Wp, bp, lab_idx, labels, eps, out,
            n_labels, dim_inner, dim_cat, dim_out);
    }
}